// ExpandingAttention_67284957659385
// MI455X (gfx1250) — compile-verified
//
#include <hip/hip_runtime.h>
#include <cmath>

#define TT 8192
#define CC 2048
#define NEGV (-1.0e30f)
#define INV_SQRT_HS 0.022097086912079608f /* 1/sqrt(2048) */

typedef __attribute__((ext_vector_type(2))) float v2f;
typedef __attribute__((ext_vector_type(8))) float v8f;

// ---------------------------------------------------------------------------
// Kernel 1: q_last[o] = sum_c x[T-1, c] * W[o, c]    (o in [0, C))
// 2048 blocks x 256 threads; coalesced over W rows; x last row stays in L2.
// ---------------------------------------------------------------------------
__global__ void __launch_bounds__(256) k_qlast(const float* __restrict__ x,
                                               const float* __restrict__ W,
                                               float* __restrict__ q) {
    const int o = blockIdx.x;
    const float* __restrict__ xr = x + (size_t)(TT - 1) * CC;
    const float* __restrict__ wr = W + (size_t)o * CC;
    float s = 0.0f;
    for (int c = threadIdx.x; c < CC; c += 256) s += xr[c] * wr[c];
    __shared__ float red[256];
    red[threadIdx.x] = s;
    __syncthreads();
    for (int k = 128; k > 0; k >>= 1) {
        if (threadIdx.x < k) red[threadIdx.x] += red[threadIdx.x + k];
        __syncthreads();
    }
    if (threadIdx.x == 0) q[o] = red[0];
}

// ---------------------------------------------------------------------------
// Kernel 2: u[c] = (sum_j W[C + j, c] * q_last[j]) / sqrt(HS)
// 8 blocks x 256 threads, thread-per-column, perfectly coalesced row sweep.
// ---------------------------------------------------------------------------
__global__ void __launch_bounds__(256) k_u(const float* __restrict__ W,
                                           const float* __restrict__ q,
                                           float* __restrict__ u) {
    const int c = blockIdx.x * 256 + threadIdx.x;
    const float* __restrict__ wk = W + (size_t)CC * CC;  // rows C .. 2C-1
    float s = 0.0f;
    #pragma unroll 4
    for (int j = 0; j < CC; ++j) s += wk[(size_t)j * CC + c] * q[j];
    u[c] = s * INV_SQRT_HS;
}

// ---------------------------------------------------------------------------
// Kernel 3: fused  att0[t] = x[t,:].u   and  vvec[t] = x[t,:].w_v
// via V_WMMA_F32_16X16X4_F32.  A = 16 rows of x (16x4 per step);
// B column 0 = u, column 1 = w_v, columns 2..15 zero.
// Branch-free B fetch: per-lane source pointer chosen once, loop-invariant
// lane mask zeroes the unused columns (v_cndmask, no EXEC churn).
// Grid: T/16 = 512 blocks x 512 threads (16 waves split K=2048).
// ---------------------------------------------------------------------------
__global__ void __launch_bounds__(512) k_scores(const float* __restrict__ x,
                                                const float* __restrict__ u,
                                                const float* __restrict__ wv,
                                                float* __restrict__ att,
                                                float* __restrict__ vvec) {
    const int tid  = threadIdx.x;
    const int wave = tid >> 5;
    const int lane = tid & 31;
    const int r    = lane & 15;   // A: row within tile; B: column N
    const int h    = lane >> 4;   // half: selects K slots {0,1} vs {2,3}
    const int t0   = blockIdx.x * 16;
    const int kbase = wave * (CC / 16) + 2 * h;   // 128-wide K slice per wave

    // Hoisted base pointers: inner loop uses immediate offsets only.
    const float* __restrict__ xp = x + (size_t)(t0 + r) * CC + kbase;
    const float* __restrict__ bsel = (r == 1) ? wv : u;   // r>=2 lanes: dummy (u)
    const float* __restrict__ bp = bsel + kbase;
    const bool live = (r < 2);

    v8f acc = {};
    #pragma unroll
    for (int i = 0; i < (CC / 16) / 4; ++i) {    // 32 WMMA steps per wave
        v2f a = *(const v2f*)(xp + i * 4);
        v2f braw = *(const v2f*)(bp + i * 4);    // unconditional, cached
        v2f b;
        b.x = live ? braw.x : 0.0f;              // loop-invariant cndmask
        b.y = live ? braw.y : 0.0f;
        acc = __builtin_amdgcn_wmma_f32_16x16x4_f32(
            /*neg_a=*/false, a, /*neg_b=*/false, b,
            /*c_mod=*/(short)0, acc, /*reuse_a=*/false, /*reuse_b=*/false);
    }

    // D layout: lane n (cols) x VGPR g: lanes 0-15 -> M=g, lanes 16-31 -> M=8+g.
    // Column 0 (att) lives in lanes 0,16; column 1 (vvec) in lanes 1,17.
    __shared__ float part[16][2][16];
    if (r < 2) {
        #pragma unroll
        for (int g = 0; g < 8; ++g) part[wave][r][h * 8 + g] = acc[g];
    }
    __syncthreads();

    if (tid < 32) {
        const int m   = tid & 15;
        const int col = tid >> 4;
        float s = 0.0f;
        #pragma unroll
        for (int w = 0; w < 16; ++w) s += part[w][col][m];
        const int t = t0 + m;
        if (col == 0) att[t] = (t == TT - 1) ? NEGV : s;
        else          vvec[t] = s;
    }
}

// ---------------------------------------------------------------------------
// Kernel 4: the 100-iteration scalar-carry softmax scan. Single block of 256
// threads; att0/vvec cached in registers (32 elements per thread).
// All threads replicate the scalar carry deterministically.
// ---------------------------------------------------------------------------
__global__ void __launch_bounds__(256) k_iter(const float* __restrict__ att,
                                              const float* __restrict__ vvec,
                                              const float* __restrict__ alpha_p,
                                              const float* __restrict__ beta_p,
                                              float* __restrict__ out) {
    const int tid = threadIdx.x;

    float att_r[32], v_r[32];
    #pragma unroll
    for (int j = 0; j < 32; ++j) {
        const int t = tid + 256 * j;
        att_r[j] = att[t];
        v_r[j]   = vvec[t];
    }

    float alpha = alpha_p[0];
    float beta  = beta_p[0];
    float k_old = 0.0f;
    float y     = 0.0f;

    __shared__ float redA[8], redB[8], redC[8];
    const float logPT = logf(0.05f);

    for (int it = 0; it < 100; ++it) {
        const float p      = alpha / (alpha + beta);
        const float kk     = logPT / logf(1.0f - p);
        const float n_xs   = fminf(ceilf(kk), (float)TT);
        const float inv_kk = 1.0f / kk;

        // ---- pass 1: global max of logits ----
        float mx = -3.0e38f;
        #pragma unroll
        for (int j = 0; j < 32; ++j) {
            const float dist = (float)(TT - 1 - (tid + 256 * j));
            const float l = (dist < n_xs) ? (att_r[j] - dist * inv_kk) : NEGV;
            mx = fmaxf(mx, l);
        }
        #pragma unroll
        for (int s = 16; s > 0; s >>= 1) mx = fmaxf(mx, __shfl_xor(mx, s, 32));
        if ((tid & 31) == 0) redA[tid >> 5] = mx;
        __syncthreads();
        mx = redA[0];
        #pragma unroll
        for (int w = 1; w < 8; ++w) mx = fmaxf(mx, redA[w]);
        __syncthreads();

        // ---- pass 2: Z, sum(e*dist), sum(e*v) ----
        float Z = 0.0f, Sd = 0.0f, Sy = 0.0f;
        #pragma unroll
        for (int j = 0; j < 32; ++j) {
            const float dist = (float)(TT - 1 - (tid + 256 * j));
            float e = 0.0f;
            if (dist < n_xs) e = expf(att_r[j] - dist * inv_kk - mx);
            Z  += e;
            Sd += e * dist;
            Sy += e * v_r[j];
        }
        #pragma unroll
        for (int s = 16; s > 0; s >>= 1) {
            Z  += __shfl_xor(Z, s, 32);
            Sd += __shfl_xor(Sd, s, 32);
            Sy += __shfl_xor(Sy, s, 32);
        }
        if ((tid & 31) == 0) {
            const int w = tid >> 5;
            redA[w] = Z; redB[w] = Sd; redC[w] = Sy;
        }
        __syncthreads();
        Z = 0.0f; Sd = 0.0f; Sy = 0.0f;
        #pragma unroll
        for (int w = 0; w < 8; ++w) { Z += redA[w]; Sd += redB[w]; Sy += redC[w]; }
        __syncthreads();

        y = Sy / Z;                       // P frozen at first breaking iter
        const bool brk = (kk > (float)TT) || (kk - k_old < 0.01f);
        if (brk) break;
        alpha += 1.0f;
        beta  += Sd / Z;
        k_old  = kk;
    }

    if (tid == 0) out[0] = y;
}

// ---------------------------------------------------------------------------
extern "C" void kernel_launch(void* const* d_in, const int* in_sizes, int n_in,
                              void* d_out, int out_size, void* d_ws, size_t ws_size,
                              hipStream_t stream) {
    const float* x     = (const float*)d_in[0];   // (1, 8192, 2048)
    const float* W     = (const float*)d_in[1];   // (4097, 2048)
    const float* alpha = (const float*)d_in[2];
    const float* beta  = (const float*)d_in[3];
    float* out = (float*)d_out;

    float* ws   = (float*)d_ws;
    float* q    = ws;                 // 2048
    float* u    = ws + 2048;          // 2048
    float* att  = ws + 4096;          // 8192
    float* vv   = ws + 4096 + 8192;   // 8192

    const float* wv = W + (size_t)2 * CC * CC;    // row 2C of W

    k_qlast <<<CC,       256, 0, stream>>>(x, W, q);
    k_u     <<<CC / 256, 256, 0, stream>>>(W, q, u);
    k_scores<<<TT / 16,  512, 0, stream>>>(x, u, wv, att, vv);
    k_iter  <<<1,        256, 0, stream>>>(att, vv, alpha, beta, out);
}